// N_Gram_Embedding_7954279432569
// MI455X (gfx1250) — compile-verified
//
#include <hip/hip_runtime.h>

// ---------------------------------------------------------------------------
// N-gram embedding-bag gather-mean for MI455X (gfx1250, wave32).
//
// One wave per (token, order). All gram rows of a token (<=12 rows x 1KB) are
// gathered into LDS with GLOBAL_LOAD_ASYNC_TO_LDS_B128 (up to 24 async 512B
// gathers in flight per wave, zero destination VGPRs), then a single
// s_wait_asynccnt 0 and an LDS read-back reduction. Tables (~62MB) stay
// L2-resident; output (50MB) is streamed with non-temporal b128 stores.
// ---------------------------------------------------------------------------

typedef float v4f __attribute__((ext_vector_type(4)));

#define NG_TOKENS   (8 * 2048)
#define MAX_ROWS    12                 // max grams per token (order 1, len 12)
#define ROW_V4      64                 // 256 floats = 64 v4f = 1024 B
#define WAVES_PER_BLOCK 4              // 128 threads

__global__ __launch_bounds__(128) void ngram_mean_async_kernel(
    const int*   __restrict__ x,
    const float* __restrict__ tab0,
    const float* __restrict__ tab1,
    const float* __restrict__ tab2,
    const float* __restrict__ tab3,
    const int*   __restrict__ ids1, const int* __restrict__ cnt1,
    const int*   __restrict__ ids2, const int* __restrict__ cnt2,
    const int*   __restrict__ ids3, const int* __restrict__ cnt3,
    float*       __restrict__ out)
{
    // 12 KB staging per wave; 4 waves -> 48 KB static LDS per block.
    __shared__ v4f smem[WAVES_PER_BLOCK * MAX_ROWS * ROW_V4];

    const int lane  = threadIdx.x & 31;
    const int w     = threadIdx.x >> 5;                 // wave in block
    const int token = blockIdx.x * WAVES_PER_BLOCK + w; // tokens per block = 4
    const int order = blockIdx.y;                       // 0..2

    // token is wave-uniform -> force x into an SGPR.
    const int xv = __builtin_amdgcn_readfirstlane(x[token]);

    float* __restrict__ dst = out + (size_t)token * 768 + (size_t)order * 256;

    if (xv < 4) {
        // Special token: copy tab0[xv, order*256 : order*256+256].
        const v4f* __restrict__ src =
            (const v4f*)(tab0 + (size_t)xv * 768 + (size_t)order * 256);
        v4f a = src[lane];
        v4f b = src[lane + 32];
        __builtin_nontemporal_store(a, (v4f*)dst + lane);
        __builtin_nontemporal_store(b, (v4f*)dst + lane + 32);
        return;
    }

    const float* __restrict__ tab;
    const int*   __restrict__ idsp;
    const int*   __restrict__ cntp;
    int L;
    if (order == 0)      { tab = tab1; idsp = ids1; cntp = cnt1; L = 12; }
    else if (order == 1) { tab = tab2; idsp = ids2; cntp = cnt2; L = 11; }
    else                 { tab = tab3; idsp = ids3; cntp = cnt3; L = 10; }

    const int  c   = __builtin_amdgcn_readfirstlane(cntp[xv]);
    const int* rid = idsp + (size_t)xv * L;

    v4f* __restrict__ mybuf = smem + w * (MAX_ROWS * ROW_V4);

    // LDS byte address for this lane's 16B slice of row 0.
    // (generic pointer = {LDS aperture, LDS offset}; low 32 bits = dsaddr)
    const unsigned lds0 = (unsigned)(uintptr_t)mybuf + (unsigned)lane * 16u;
    const unsigned long long base = (unsigned long long)(uintptr_t)tab;

    // Issue ALL gather traffic for this token asynchronously into LDS:
    // 2 instructions per 1KB row (offset:512 advances both LDS and global
    // addresses per the ISA), up to 24 async ops in flight, no dest VGPRs.
    for (int j = 0; j < c; ++j) {
        const int      g    = __builtin_amdgcn_readfirstlane(rid[j]);
        const unsigned goff = (unsigned)g * 1024u + (unsigned)lane * 16u;
        const unsigned loff = lds0 + (unsigned)j * 1024u;
        asm volatile("global_load_async_to_lds_b128 %0, %1, %2"
                     :: "v"(loff), "v"(goff), "s"(base) : "memory");
        asm volatile("global_load_async_to_lds_b128 %0, %1, %2 offset:512"
                     :: "v"(loff), "v"(goff), "s"(base) : "memory");
    }

    // Single wait for the whole row set, then reduce out of LDS.
    asm volatile("s_wait_asynccnt 0x0" ::: "memory");

    v4f acc0 = (v4f)0.0f;
    v4f acc1 = (v4f)0.0f;
    int j = 0;
    for (; j + 2 <= c; j += 2) {
        v4f a0 = mybuf[(j + 0) * ROW_V4 + lane];
        v4f a1 = mybuf[(j + 0) * ROW_V4 + lane + 32];
        v4f b0 = mybuf[(j + 1) * ROW_V4 + lane];
        v4f b1 = mybuf[(j + 1) * ROW_V4 + lane + 32];
        acc0 += a0 + b0;
        acc1 += a1 + b1;
    }
    if (j < c) {
        acc0 += mybuf[j * ROW_V4 + lane];
        acc1 += mybuf[j * ROW_V4 + lane + 32];
    }

    const float inv = 1.0f / (float)c;   // c >= 1 always
    acc0 *= inv;
    acc1 *= inv;

    // Non-temporal: don't let the 50MB output stream evict the L2-resident
    // embedding tables.
    __builtin_nontemporal_store(acc0, (v4f*)dst + lane);
    __builtin_nontemporal_store(acc1, (v4f*)dst + lane + 32);
}

extern "C" void kernel_launch(void* const* d_in, const int* in_sizes, int n_in,
                              void* d_out, int out_size, void* d_ws, size_t ws_size,
                              hipStream_t stream) {
    // setup_inputs() order:
    //   0:x  1:tab0  2:tab1  3:tab2  4:tab3
    //   5:ids1 6:cnt1 7:ids2 8:cnt2 9:ids3 10:cnt3
    const int*   x    = (const int*)  d_in[0];
    const float* tab0 = (const float*)d_in[1];
    const float* tab1 = (const float*)d_in[2];
    const float* tab2 = (const float*)d_in[3];
    const float* tab3 = (const float*)d_in[4];
    const int*   ids1 = (const int*)  d_in[5];
    const int*   cnt1 = (const int*)  d_in[6];
    const int*   ids2 = (const int*)  d_in[7];
    const int*   cnt2 = (const int*)  d_in[8];
    const int*   ids3 = (const int*)  d_in[9];
    const int*   cnt3 = (const int*)  d_in[10];
    float*       out  = (float*)      d_out;

    dim3 grid(NG_TOKENS / WAVES_PER_BLOCK, 3, 1);   // (4096, 3): order on y
    dim3 block(WAVES_PER_BLOCK * 32, 1, 1);         // 128 threads = 4 wave32s

    ngram_mean_async_kernel<<<grid, block, 0, stream>>>(
        x, tab0, tab1, tab2, tab3,
        ids1, cnt1, ids2, cnt2, ids3, cnt3, out);
}